// Loss_26886495273742
// MI455X (gfx1250) — compile-verified
//
#include <hip/hip_runtime.h>

typedef float v2f __attribute__((ext_vector_type(2)));
typedef float v8f __attribute__((ext_vector_type(8)));

#define MARGIN_F 0.5f
#define EPS_F    1e-5f
#define D_K      256
#define A_PAD    260   // (m*260 + k) % 64 hits 16 distinct banks across lanes 0..15

#define WMMA_F32(a, b, c) \
    __builtin_amdgcn_wmma_f32_16x16x4_f32(false, (a), false, (b), (short)0, (c), false, false)

// Fused  sim = F * F^T  tile GEMM (fp32 WMMA 16x16x4) + masked row reductions.
// Block: 256 threads (8 waves), 32 rows.  Each wave: 2 row strips x 2 col
// subtiles (32x32 of sim per iteration), K software-pipelined in 16-wide
// chunks with double-buffered B fragments in registers.
// mode 0: accumulate neg-sum over identity columns          -> negOut
// mode 1: accumulate pos-sum and neg-sum over gathered cols -> posOut, negOut
__global__ __launch_bounds__(256) void sim_rowsum_kernel(
    const float* __restrict__ feature,
    const int*   __restrict__ target,
    const int*   __restrict__ colIdx,   // nullptr => identity columns
    float* __restrict__ posOut,
    float* __restrict__ negOut,
    int n, int d, int ncols, int mode)
{
    __shared__ float Ash[32 * A_PAD];
    __shared__ int   tRow[32];
    __shared__ float shPos[32];
    __shared__ float shNeg[32];

    const int tid  = threadIdx.x;
    const int wave = tid >> 5;
    const int lane = tid & 31;
    const int m    = lane & 15;
    const int hi   = lane >> 4;      // which 16-lane half
    const int koff = hi * 2;         // K sub-offset for 16x16x4 fragments
    const int row0 = blockIdx.x * 32;

    // Stage the 32-row A strip into LDS (row-major, padded stride).
    for (int idx = tid; idx < 32 * D_K; idx += 256) {
        int r = idx >> 8;            // / 256
        int k = idx & 255;
        Ash[r * A_PAD + k] = feature[(size_t)(row0 + r) * d + k];
    }
    if (tid < 32) {
        tRow[tid]  = target[row0 + tid];
        shPos[tid] = 0.0f;
        shNeg[tid] = 0.0f;
    }
    __syncthreads();

    const int aBase0 = m * A_PAD + koff;             // row strip 0
    const int aBase1 = (16 + m) * A_PAD + koff;      // row strip 1

    float posAcc[2][8], negAcc[2][8];
    #pragma unroll
    for (int st = 0; st < 2; ++st)
        #pragma unroll
        for (int r = 0; r < 8; ++r) { posAcc[st][r] = 0.0f; negAcc[st][r] = 0.0f; }

    const v8f vzero = {};
    const int nTiles32 = ncols >> 5;          // 32 columns per wave-iteration
    for (int t = wave; t < nTiles32; t += 8) {
        const int c0 = t << 5;

        // Column gather: B row pointers (koff folded in) + column targets.
        const float* bp0;
        const float* bp1;
        int tCol[2];
        {
            int cn0 = c0 + m,      cg0 = colIdx ? colIdx[cn0] : cn0;
            int cn1 = c0 + 16 + m, cg1 = colIdx ? colIdx[cn1] : cn1;
            bp0 = feature + (size_t)cg0 * d + koff;
            bp1 = feature + (size_t)cg1 * d + koff;
            tCol[0] = target[cg0];
            tCol[1] = target[cg1];
        }

        v8f acc[2][2];
        #pragma unroll
        for (int st = 0; st < 2; ++st)
            #pragma unroll
            for (int s = 0; s < 2; ++s) acc[st][s] = vzero;

        v2f b0[2][4], b1[2][4];   // double-buffered B fragments (static idx only)

        auto loadB = [&](v2f (&bb)[2][4], int k) {
            #pragma unroll
            for (int j = 0; j < 4; ++j) {
                bb[0][j] = *(const v2f*)(bp0 + k + 4 * j);
                bb[1][j] = *(const v2f*)(bp1 + k + 4 * j);
            }
        };
        auto compute = [&](v2f (&bb)[2][4], int kc) {
            v2f a0[4], a1[4];
            #pragma unroll
            for (int j = 0; j < 4; ++j) {
                a0[j] = *(const v2f*)&Ash[aBase0 + kc + 4 * j];
                a1[j] = *(const v2f*)&Ash[aBase1 + kc + 4 * j];
            }
            #pragma unroll
            for (int j = 0; j < 4; ++j) {
                acc[0][0] = WMMA_F32(a0[j], bb[0][j], acc[0][0]);
                acc[0][1] = WMMA_F32(a0[j], bb[1][j], acc[0][1]);
                acc[1][0] = WMMA_F32(a1[j], bb[0][j], acc[1][0]);
                acc[1][1] = WMMA_F32(a1[j], bb[1][j], acc[1][1]);
            }
        };

        // Software pipeline over K: load chunk k+16 while computing chunk k.
        loadB(b0, 0);
        for (int kc = 0; kc < d; kc += 32) {
            loadB(b1, kc + 16);
            compute(b0, kc);
            if (kc + 32 < d) loadB(b0, kc + 32);
            compute(b1, kc + 16);
        }

        // Epilogue: element (strip st, subtile s, reg r, this lane) is
        //   sim[row0 + 16*st + r + 8*hi][c0 + 16*s + m]
        #pragma unroll
        for (int st = 0; st < 2; ++st) {
            #pragma unroll
            for (int s = 0; s < 2; ++s) {
                int tc = tCol[s];
                #pragma unroll
                for (int r = 0; r < 8; ++r) {
                    float sv = acc[st][s][r];
                    int   tr = tRow[16 * st + r + 8 * hi];
                    bool diff = (tr != tc);
                    if (diff && sv > MARGIN_F) negAcc[st][r] += sv;
                    if (mode == 1 && !diff && sv < (1.0f - EPS_F))
                        posAcc[st][r] += (1.0f - sv);
                }
            }
        }
    }

    // Reduce across the 16 lanes of each half (xor masks stay within halves),
    // then combine across waves with LDS float atomics.
    #pragma unroll
    for (int st = 0; st < 2; ++st) {
        #pragma unroll
        for (int r = 0; r < 8; ++r) {
            float pn = posAcc[st][r], ng = negAcc[st][r];
            #pragma unroll
            for (int off = 8; off >= 1; off >>= 1) {
                pn += __shfl_xor(pn, off, 32);
                ng += __shfl_xor(ng, off, 32);
            }
            posAcc[st][r] = pn; negAcc[st][r] = ng;
        }
    }
    if (m == 0) {
        #pragma unroll
        for (int st = 0; st < 2; ++st) {
            #pragma unroll
            for (int r = 0; r < 8; ++r) {
                atomicAdd(&shNeg[16 * st + r + 8 * hi], negAcc[st][r]);
                if (mode == 1) atomicAdd(&shPos[16 * st + r + 8 * hi], posAcc[st][r]);
            }
        }
    }
    __syncthreads();

    if (tid < 32) {
        negOut[row0 + tid] = shNeg[tid];
        if (mode == 1) posOut[row0 + tid] = shPos[tid];
    }
}

__global__ __launch_bounds__(256) void finalize_kernel(
    const float* __restrict__ pos,
    const float* __restrict__ negAll,
    const float* __restrict__ negNew,
    float* __restrict__ out,
    int n, int n_new, float alpha)
{
    __shared__ float red[256];
    const float stale = pos[n_new - 1];
    float acc = 0.0f;
    for (int i = threadIdx.x; i < n; i += 256) {
        bool isNew = i < n_new;
        float p = isNew ? pos[i]    : stale;
        float g = isNew ? negAll[i] : negNew[i];
        acc += alpha * p + (1.0f - alpha) * g;
    }
    red[threadIdx.x] = acc;
    __syncthreads();
    for (int s = 128; s > 0; s >>= 1) {
        if (threadIdx.x < s) red[threadIdx.x] += red[threadIdx.x + s];
        __syncthreads();
    }
    if (threadIdx.x == 0) out[0] = red[0] / (float)n;
}

extern "C" void kernel_launch(void* const* d_in, const int* in_sizes, int n_in,
                              void* d_out, int out_size, void* d_ws, size_t ws_size,
                              hipStream_t stream) {
    const float* feature = (const float*)d_in[0];
    const int*   target  = (const int*)d_in[1];
    const int*   new_idx = (const int*)d_in[2];

    const int n     = in_sizes[1];
    const int d     = in_sizes[0] / in_sizes[1];
    const int n_new = in_sizes[2];
    const int n_old = in_sizes[3];
    const float alpha = (n_old != 0) ? 0.9f : 0.5f;

    float* pos    = (float*)d_ws;        // n floats
    float* negAll = pos + n;             // n floats (rows < n_new used)
    float* negNew = negAll + n;          // n floats

    // Pass 1: rows [0, n_new) x ALL columns -> neg_all
    sim_rowsum_kernel<<<n_new / 32, 256, 0, stream>>>(
        feature, target, nullptr, nullptr, negAll, n, d, n, 0);

    // Pass 2: ALL rows x new columns (gathered) -> pos_loss_new, neg_newcols
    sim_rowsum_kernel<<<n / 32, 256, 0, stream>>>(
        feature, target, new_idx, pos, negNew, n, d, n_new, 1);

    // Blend with stale-pos rule and take the mean.
    finalize_kernel<<<1, 256, 0, stream>>>(
        pos, negAll, negNew, (float*)d_out, n, n_new, alpha);
}